// GAE_23441931501893
// MI455X (gfx1250) — compile-verified
//
#include <hip/hip_runtime.h>
#include <hip/hip_bf16.h>

typedef __attribute__((ext_vector_type(2))) float v2f;
typedef __attribute__((ext_vector_type(8))) float v8f;

#define LDS_PITCH 132  // 128 + 4 pad -> conflict-free column reads (64 banks)

// ---------------------------------------------------------------------------
// C[M x NOUT] = act(A[M x 128]) @ W[128 x NOUT]   via V_WMMA_F32_16X16X4_F32
// One block per 16-row M tile, one wave32 per 16-col N tile.
// A fragment (16x4 f32): lanes 0-15 hold M=lane, K=k0..k0+1 (VGPR0/1);
//                        lanes 16-31 hold K=k0+2..k0+3.
// C/D (16x16 f32): VGPR i -> M=i (lanes 0-15), M=i+8 (lanes 16-31), N=lane%16.
// ---------------------------------------------------------------------------
template <int NOUT, bool RELU>
__global__ __launch_bounds__(NOUT * 2) void gcn_gemm_wmma(
    const float* __restrict__ A, const float* __restrict__ W,
    float* __restrict__ C, int M) {
  __shared__ float tile[16 * LDS_PITCH];
  const int m0 = blockIdx.x * 16;
  const bool full = (m0 + 16 <= M);  // M % 16 == 0 in practice -> always true

  // Cooperative, coalesced load of the 16x128 A tile (optional fused ReLU).
  if (full) {
    for (int i = threadIdx.x; i < 16 * 128; i += blockDim.x) {
      const int r = i >> 7, c = i & 127;
      float v = A[(m0 + r) * 128 + c];
      if (RELU) v = fmaxf(v, 0.0f);
      tile[r * LDS_PITCH + c] = v;
    }
  } else {
    for (int i = threadIdx.x; i < 16 * 128; i += blockDim.x) {
      const int r = i >> 7, c = i & 127;
      const int row = m0 + r;
      float v = 0.0f;
      if (row < M) v = A[row * 128 + c];
      if (RELU) v = fmaxf(v, 0.0f);
      tile[r * LDS_PITCH + c] = v;
    }
  }
  __syncthreads();

  const int wave = threadIdx.x >> 5;
  const int lane = threadIdx.x & 31;
  const int half = lane >> 4;   // 0: lanes 0-15, 1: lanes 16-31
  const int l16  = lane & 15;
  const int n    = wave * 16 + l16;

  v8f acc = {};
#pragma unroll 4
  for (int k0 = 0; k0 < 128; k0 += 4) {
    const int ka = k0 + (half << 1);
    v2f a, b;
    a.x = tile[l16 * LDS_PITCH + ka];
    a.y = tile[l16 * LDS_PITCH + ka + 1];
    b.x = W[(ka + 0) * NOUT + n];
    b.y = W[(ka + 1) * NOUT + n];
    // 8 args: (neg_a, A, neg_b, B, c_mod, C, reuse_a, reuse_b)
    acc = __builtin_amdgcn_wmma_f32_16x16x4_f32(
        false, a, false, b, (short)0, acc, false, false);
  }

  if (full) {
#pragma unroll
    for (int i = 0; i < 8; ++i) {
      C[(m0 + i + half * 8) * NOUT + n] = acc[i];
    }
  } else {
#pragma unroll
    for (int i = 0; i < 8; ++i) {
      const int row = m0 + i + half * 8;
      if (row < M) C[row * NOUT + n] = acc[i];
    }
  }
}

// ---------------------------------------------------------------------------
// Degree / normalization helpers.  deg starts at 1 (self-loop), so deg > 0.
// ---------------------------------------------------------------------------
__global__ void deg_init(float* __restrict__ deg, int N) {
  const int i = blockIdx.x * blockDim.x + threadIdx.x;
  if (i < N) deg[i] = 1.0f;
}

__global__ void deg_count(const int* __restrict__ dst, float* __restrict__ deg,
                          int E) {
  const int e = blockIdx.x * blockDim.x + threadIdx.x;
  if (e < E) atomicAdd(&deg[dst[e]], 1.0f);
}

__global__ void deg_finalize(float* __restrict__ deg, int N) {
  const int i = blockIdx.x * blockDim.x + threadIdx.x;
  if (i < N) deg[i] = rsqrtf(deg[i]);
}

// out[i][j] = b[j]  (NOUT is a power of two)
template <int NOUT>
__global__ void bias_init(float* __restrict__ out, const float* __restrict__ b,
                          int total) {
  const int i = blockIdx.x * blockDim.x + threadIdx.x;
  if (i < total) out[i] = b[i & (NOUT - 1)];
}

// ---------------------------------------------------------------------------
// Scatter-add message pass: out[d] += h[s] * dinv[s]*dinv[d] over E real edges
// plus N self-loops.  NF/4 lanes cooperate on one edge (float4 per lane);
// h rows hit L2 (h fits entirely in the 192 MB L2), atomics resolve in L2.
// ---------------------------------------------------------------------------
template <int NF>
__global__ void gcn_scatter(const float* __restrict__ h,
                            const float* __restrict__ dinv,
                            const int* __restrict__ src,
                            const int* __restrict__ dst,
                            float* __restrict__ out, int E, int N) {
  constexpr int LPE = NF / 4;
  const int tid  = blockIdx.x * blockDim.x + threadIdx.x;
  const int edge = tid / LPE;
  const int lane = tid % LPE;
  if (edge >= E + N) return;

  int s, d;
  if (edge < E) {
    s = src[edge];
    d = dst[edge];
  } else {
    s = d = edge - E;  // self loop
  }
  const float norm = dinv[s] * dinv[d];
  const float4 hv = *(const float4*)(h + s * NF + lane * 4);
  float* o = out + d * NF + lane * 4;
  atomicAdd(o + 0, hv.x * norm);
  atomicAdd(o + 1, hv.y * norm);
  atomicAdd(o + 2, hv.z * norm);
  atomicAdd(o + 3, hv.w * norm);
}

// ---------------------------------------------------------------------------
// Decoder: logits[i] = dot(z[e0[i]], z[e1[i]]) over 64 features.
// One wave32 per edge; float2 per lane; shfl_xor tree reduction.
// ---------------------------------------------------------------------------
__global__ void decode_dot(const float* __restrict__ z,
                           const int* __restrict__ pos,
                           const int* __restrict__ neg,
                           float* __restrict__ logits, int E) {
  const int tid  = blockIdx.x * blockDim.x + threadIdx.x;
  const int i    = tid >> 5;
  const int lane = tid & 31;
  if (i >= 2 * E) return;

  int a, b;
  if (i < E) {
    a = pos[i];
    b = pos[E + i];
  } else {
    a = neg[i - E];
    b = neg[E + (i - E)];
  }
  const float2 za = ((const float2*)z)[a * 32 + lane];
  const float2 zb = ((const float2*)z)[b * 32 + lane];
  float p = za.x * zb.x + za.y * zb.y;
#pragma unroll
  for (int off = 16; off > 0; off >>= 1) p += __shfl_xor(p, off, 32);
  if (lane == 0) logits[i] = p;
}

// ---------------------------------------------------------------------------
extern "C" void kernel_launch(void* const* d_in, const int* in_sizes, int n_in,
                              void* d_out, int out_size, void* d_ws,
                              size_t ws_size, hipStream_t stream) {
  const float* x   = (const float*)d_in[0];
  const int*   pos = (const int*)d_in[1];
  const int*   neg = (const int*)d_in[2];
  const float* W1  = (const float*)d_in[3];
  const float* b1  = (const float*)d_in[4];
  const float* W2  = (const float*)d_in[5];
  const float* b2  = (const float*)d_in[6];

  const int H1 = in_sizes[4];               // 128
  const int H2 = in_sizes[6];               // 64
  const int F  = in_sizes[3] / H1;          // 128
  const int Nn = in_sizes[0] / F;           // 50000
  const int E  = in_sizes[1] / 2;           // 500000
  const int* src = pos;                     // pos_edge_index[0]
  const int* dst = pos + E;                 // pos_edge_index[1]

  // Workspace layout (floats); h1 region recycled for h2 + out2.
  float* ws   = (float*)d_ws;
  float* dinv = ws;                               // N
  float* h1   = ws + Nn;                          // N x 128
  float* out1 = h1 + (size_t)Nn * H1;             // N x 128
  float* h2   = h1;                               // N x 64 (reuse)
  float* out2 = h1 + (size_t)Nn * H2;             // N x 64 (reuse)

  const int TB = 256;

  // Degree normalization: dinv = rsqrt(in_degree + 1)
  deg_init<<<(Nn + TB - 1) / TB, TB, 0, stream>>>(dinv, Nn);
  deg_count<<<(E + TB - 1) / TB, TB, 0, stream>>>(dst, dinv, E);
  deg_finalize<<<(Nn + TB - 1) / TB, TB, 0, stream>>>(dinv, Nn);

  // Layer 1: h1 = x @ W1 ; out1 = b1 + scatter(norm * h1)
  gcn_gemm_wmma<128, false><<<(Nn + 15) / 16, 256, 0, stream>>>(x, W1, h1, Nn);
  bias_init<128><<<(Nn * 128 + TB - 1) / TB, TB, 0, stream>>>(out1, b1,
                                                              Nn * 128);
  {
    const int work = (E + Nn) * 32;  // 32 lanes per edge (128 feats / 4)
    gcn_scatter<128><<<(work + TB - 1) / TB, TB, 0, stream>>>(h1, dinv, src,
                                                              dst, out1, E, Nn);
  }

  // Layer 2: h2 = relu(out1) @ W2 ; out2 = b2 + scatter(norm * h2)
  gcn_gemm_wmma<64, true><<<(Nn + 15) / 16, 128, 0, stream>>>(out1, W2, h2, Nn);
  bias_init<64><<<(Nn * 64 + TB - 1) / TB, TB, 0, stream>>>(out2, b2, Nn * 64);
  {
    const int work = (E + Nn) * 16;  // 16 lanes per edge (64 feats / 4)
    gcn_scatter<64><<<(work + TB - 1) / TB, TB, 0, stream>>>(h2, dinv, src, dst,
                                                             out2, E, Nn);
  }

  // Decoder over pos + neg edges
  {
    const int work = 2 * E * 32;
    decode_dot<<<(work + TB - 1) / TB, TB, 0, stream>>>(out2, pos, neg,
                                                        (float*)d_out, E);
  }
}